// Model_49538152792439
// MI455X (gfx1250) — compile-verified
//
#include <hip/hip_runtime.h>

// BEV pooling (bev_pool_v2): out[ranks_bev[p], :] += depth[ranks_depth[p]] * feat[ranks_feat[p], :]
// Memory-bound gather/scatter (240 MFLOP vs ~0.5 GB) -> no WMMA; optimize data movement.
// Block stages a 2048-point index tile into LDS via GLOBAL_LOAD_ASYNC_TO_LDS_B32
// (ASYNCcnt path), then 16-lane slots run-accumulate over sorted ranks_bev and
// flush with global_atomic_add_f32.

#define C_CH   80      // channels
#define GSIZE  16      // lanes per slot (C_CH = GSIZE * CPL)
#define CPL    5       // channels per lane
#define CHUNK  128     // points per slot
#define SLOTS  16      // slots per 256-thread block
#define TILE_P (CHUNK * SLOTS)   // 2048 points staged per block

// ---- gfx1250 async-to-LDS detection -------------------------------------
#if defined(__has_builtin)
#  if __has_builtin(__builtin_amdgcn_global_load_async_to_lds_b32)
#    define HAVE_ASYNC_LDS 1
#  endif
#  if __has_builtin(__builtin_amdgcn_s_wait_asynccnt)
#    define HAVE_WAIT_ASYNC 1
#  endif
#endif

#if defined(HAVE_ASYNC_LDS)
typedef __attribute__((address_space(1))) int GInt;   // global
typedef __attribute__((address_space(3))) int LInt;   // LDS

static __device__ __forceinline__ void async_ld_b32(const int* g, int* l) {
    // signature per hipcc diagnostic: (global int*, lds int*, imm offset, imm cpol)
    __builtin_amdgcn_global_load_async_to_lds_b32((GInt*)(int*)g, (LInt*)l, 0, 0);
}
static __device__ __forceinline__ void wait_async0() {
#  if defined(HAVE_WAIT_ASYNC)
    __builtin_amdgcn_s_wait_asynccnt(0);
#  else
    asm volatile("s_wait_asynccnt 0x0" ::: "memory");
#  endif
}
#endif
// -------------------------------------------------------------------------

__global__ __launch_bounds__(256) void bev_pool(
    const float* __restrict__ depth,        // [498432]
    const float* __restrict__ feat,         // [4224, 80]
    const int*   __restrict__ ranks_depth,  // [P]
    const int*   __restrict__ ranks_feat,   // [P]
    const int*   __restrict__ ranks_bev,    // [P], sorted
    float*       __restrict__ out,          // [40000, 80]
    int P)
{
    __shared__ int2  s_idx[TILE_P];   // .x = ranks_bev, .y = ranks_feat
    __shared__ float s_d  [TILE_P];   // pre-gathered depth weight

    const int tid      = threadIdx.x;
    const int block_p0 = blockIdx.x * TILE_P;
    const int block_n  = (P - block_p0 < TILE_P) ? (P - block_p0) : TILE_P;

    // ---- Stage tile: rb/rf async to LDS, depth gathered through VGPRs ----
    for (int k = tid; k < block_n; k += blockDim.x) {
#if defined(HAVE_ASYNC_LDS)
        async_ld_b32(ranks_bev  + block_p0 + k, &s_idx[k].x);  // ASYNCcnt path
        async_ld_b32(ranks_feat + block_p0 + k, &s_idx[k].y);
#else
        s_idx[k].x = ranks_bev [block_p0 + k];
        s_idx[k].y = ranks_feat[block_p0 + k];
#endif
        const int rd = ranks_depth[block_p0 + k];              // coalesced
        s_d[k] = depth[rd];                                    // random, L2-resident
    }
#if defined(HAVE_ASYNC_LDS)
    wait_async0();
#endif
    __syncthreads();

    // ---- Per-slot run accumulation over sorted ranks_bev ----
    const int slot = tid / GSIZE;
    const int sub  = tid & (GSIZE - 1);
    const int l0   = slot * CHUNK;
    const int l1   = (l0 + CHUNK < block_n) ? (l0 + CHUNK) : block_n;

    float acc[CPL];
#pragma unroll
    for (int j = 0; j < CPL; ++j) acc[j] = 0.f;
    int cur_rb = -1;

    for (int l = l0; l < l1; ++l) {
        const int2  idx = s_idx[l];   // one ds_load_b64 broadcast per point
        const float d   = s_d[l];

        // Prefetch the feature row 8 points ahead (3 cachelines via lanes 0..2).
        if (sub < 3 && l + 8 < l1) {
            const float* fa = feat + (size_t)s_idx[l + 8].y * C_CH + sub * 32;
            __builtin_prefetch(fa, 0, 0);
        }

        if (idx.x != cur_rb) {
            if (cur_rb >= 0) {
                float* o = out + (size_t)cur_rb * C_CH + sub;
#pragma unroll
                for (int j = 0; j < CPL; ++j) {
                    atomicAdd(o + j * GSIZE, acc[j]);
                    acc[j] = 0.f;
                }
            }
            cur_rb = idx.x;
        }

        const float* f = feat + (size_t)idx.y * C_CH + sub;    // 5 x 64B slices
#pragma unroll
        for (int j = 0; j < CPL; ++j)
            acc[j] = fmaf(d, f[j * GSIZE], acc[j]);
    }

    if (cur_rb >= 0) {
        float* o = out + (size_t)cur_rb * C_CH + sub;
#pragma unroll
        for (int j = 0; j < CPL; ++j)
            atomicAdd(o + j * GSIZE, acc[j]);
    }
}

__global__ __launch_bounds__(256) void bev_zero(float* __restrict__ out, int n) {
    int i = (blockIdx.x * blockDim.x + threadIdx.x) * 4;
    if (i + 3 < n) {
        float4 z = {0.f, 0.f, 0.f, 0.f};
        *reinterpret_cast<float4*>(out + i) = z;
    } else {
        for (; i < n; ++i) out[i] = 0.f;
    }
}

extern "C" void kernel_launch(void* const* d_in, const int* in_sizes, int n_in,
                              void* d_out, int out_size, void* d_ws, size_t ws_size,
                              hipStream_t stream) {
    // Inputs: 0 depth(f32), 1 feat(f32), 2 ranks_depth(i32), 3 ranks_feat(i32),
    //         4 ranks_bev(i32), 5 interval_starts(unused), 6 interval_lengths(unused),
    //         7 total_bev(scalar)
    const float* depth       = (const float*)d_in[0];
    const float* feat        = (const float*)d_in[1];
    const int*   ranks_depth = (const int*)d_in[2];
    const int*   ranks_feat  = (const int*)d_in[3];
    const int*   ranks_bev   = (const int*)d_in[4];
    float*       out         = (float*)d_out;

    const int P = in_sizes[2];

    {   // Zero the poisoned output accumulator.
        const int threads = 256;
        const int blocks  = (out_size + threads * 4 - 1) / (threads * 4);
        bev_zero<<<blocks, threads, 0, stream>>>(out, out_size);
    }

    if (P > 0) {
        const int blocks = (P + TILE_P - 1) / TILE_P;
        bev_pool<<<blocks, 256, 0, stream>>>(depth, feat, ranks_depth,
                                             ranks_feat, ranks_bev, out, P);
    }
}